// AGAT_29600914604432
// MI455X (gfx1250) — compile-verified
//
#include <hip/hip_runtime.h>
#include <math.h>

#define NN 16384
#define EE 262144
#define TT 4
#define DD 64
#define LL 2

typedef float v2f __attribute__((ext_vector_type(2)));
typedef float v8f __attribute__((ext_vector_type(8)));

// ---------- helpers: order-preserving float <-> uint for atomicMax ----------
__device__ __forceinline__ unsigned f2ord(float f) {
    unsigned u = __float_as_uint(f);
    return (u & 0x80000000u) ? ~u : (u | 0x80000000u);
}
__device__ __forceinline__ float ord2f(unsigned u) {
    return (u & 0x80000000u) ? __uint_as_float(u & 0x7fffffffu)
                             : __uint_as_float(~u);
}

// ---------- init: h[t,n,d] = x[n,d]; zero-fill; small copies ----------
__global__ void k_bcast_h(const float* __restrict__ x, float* __restrict__ h) {
    int i = blockIdx.x * blockDim.x + threadIdx.x;       // < TT*NN*DD
    h[i] = x[i & (NN * DD - 1)];                         // NN*DD is a power of 2
}

__global__ void k_copy256(const float* __restrict__ src, float* __restrict__ dst) {
    int i = threadIdx.x;                                 // TT*DD == 256
    dst[i] = src[i];
}

__global__ void k_fill0(float* __restrict__ p, int n) {
    int i = blockIdx.x * blockDim.x + threadIdx.x;
    if (i < n) p[i] = 0.0f;
}

__global__ void k_relu(float* __restrict__ p, int n) {
    int i = blockIdx.x * blockDim.x + threadIdx.x;
    if (i < n) p[i] = fmaxf(p[i], 0.0f);
}

// ---------- per-layer small prep: g = tg @ ef_in^T ; ef_out = ef_in @ wr ; vg = sigmoid(ef_out)
__global__ void k_prep(const float* __restrict__ th,      // T x 3D (this layer)
                       const float* __restrict__ ef_in,   // T x D
                       const float* __restrict__ wr,      // D x D
                       float* __restrict__ g,             // T x T
                       float* __restrict__ ef_out,        // T x D
                       float* __restrict__ vg) {          // T x D
    int tid = threadIdx.x;                                // 256 threads
    if (tid < TT * TT) {
        int a = tid / TT, b = tid % TT;
        float s = 0.0f;
        for (int k = 0; k < DD; ++k) s += th[a * 3 * DD + k] * ef_in[b * DD + k];
        g[tid] = s;
    }
    int t = tid / DD, d = tid % DD;
    float s = 0.0f;
    for (int k = 0; k < DD; ++k) s += ef_in[t * DD + k] * wr[k * DD + d];
    ef_out[tid] = s;
    vg[tid] = 1.0f / (1.0f + __expf(-s));
}

// ---------- node scores: a_i[t,n] = h[t,n]·thi[t]; a_j[t,n] = h[t,n]·thj[t]
__global__ void k_node_scores(const float* __restrict__ h,
                              const float* __restrict__ th,
                              float* __restrict__ ai, float* __restrict__ aj) {
    int id = blockIdx.x * blockDim.x + threadIdx.x;       // < TT*NN
    int t = id / NN;
    const float* hp = h + (size_t)id * DD;
    const float* ti = th + t * 3 * DD + DD;
    const float* tj = th + t * 3 * DD + 2 * DD;
    float si = 0.0f, sj = 0.0f;
    #pragma unroll 8
    for (int k = 0; k < DD; ++k) { float hv = hp[k]; si += hv * ti[k]; sj += hv * tj[k]; }
    ai[id] = si; aj[id] = sj;
}

// ---------- WMMA GEMM: hw[t] = h[t] (N x D) @ we (D x D), fp32, 16x16x4 ----------
// block = 128 threads (4 waves); grid = (N/16, T). Wave w handles column tile n0=w*16.
__global__ void k_gemm_wmma(const float* __restrict__ h,
                            const float* __restrict__ we,
                            float* __restrict__ hw) {
    __shared__ float lw[DD * DD];                         // 16 KB: stage we in LDS
    int tid = threadIdx.x;
    for (int i = tid; i < DD * DD / 4; i += 128)
        ((float4*)lw)[i] = ((const float4*)we)[i];
    __syncthreads();

    int wave = tid >> 5, lane = tid & 31;
    int lhalf = lane >> 4;                                // 0: lanes 0-15, 1: lanes 16-31
    int l15   = lane & 15;
    int m0 = blockIdx.x * 16;
    int t  = blockIdx.y;
    int n0 = wave * 16;

    const float* hb = h + ((size_t)t * NN + m0) * DD;
    v8f c = {};
    #pragma unroll
    for (int k0 = 0; k0 < DD; k0 += 4) {
        // A 16x4 fragment: lane half selects K pair {k0,k0+1} vs {k0+2,k0+3}, M = l15
        int kb = k0 + (lhalf << 1);
        v2f a, b;
        a.x = hb[l15 * DD + kb];
        a.y = hb[l15 * DD + kb + 1];
        // B 4x16 fragment: same K pair, N = n0 + l15 (row striped across lanes per VGPR)
        b.x = lw[kb * DD + n0 + l15];
        b.y = lw[(kb + 1) * DD + n0 + l15];
        c = __builtin_amdgcn_wmma_f32_16x16x4_f32(false, a, false, b, (short)0, c, false, false);
    }
    // C/D layout: VGPR v -> M = v + 8*lhalf, N = n0 + l15
    float* ob = hw + ((size_t)t * NN + m0) * DD + n0;
    #pragma unroll
    for (int v = 0; v < 8; ++v)
        ob[(v + lhalf * 8) * DD + l15] = c[v];
}

// ---------- edge logits + segment max (ordered-uint atomicMax) ----------
__global__ void k_edge_r(const int* __restrict__ row, const int* __restrict__ col,
                         const int* __restrict__ et,
                         const float* __restrict__ g, const float* __restrict__ ai,
                         const float* __restrict__ aj,
                         float* __restrict__ re, unsigned* __restrict__ smax) {
    int e = blockIdx.x * blockDim.x + threadIdx.x;
    if (e >= EE) return;
    int r = row[e], c = col[e], ty = et[e];
    #pragma unroll
    for (int t = 0; t < TT; ++t) {
        float v = g[t * TT + ty] + ai[t * NN + r] + aj[t * NN + c];
        re[(size_t)t * EE + e] = v;
        atomicMax(&smax[t * NN + r], f2ord(v));
    }
}

// ---------- exp(r - m) + segment sum ----------
__global__ void k_edge_exp(const int* __restrict__ row,
                           const unsigned* __restrict__ smax,
                           float* __restrict__ re, float* __restrict__ ssum) {
    int e = blockIdx.x * blockDim.x + threadIdx.x;
    if (e >= EE) return;
    int r = row[e];
    #pragma unroll
    for (int t = 0; t < TT; ++t) {
        float m  = ord2f(smax[t * NN + r]);
        float ex = __expf(re[(size_t)t * EE + e] - m);
        re[(size_t)t * EE + e] = ex;
        atomicAdd(&ssum[t * NN + r], ex);
    }
}

// ---------- aggregation: out[t,row] += (e/s) * vg[t] * hw[t,col] ----------
// one wave per (edge, type); lane covers d = lane and d = lane+32
__global__ void k_agg(const int* __restrict__ row, const int* __restrict__ col,
                      const float* __restrict__ re, const float* __restrict__ ssum,
                      const float* __restrict__ vg, const float* __restrict__ hw,
                      float* __restrict__ out) {
    int gid  = blockIdx.x * 8 + (threadIdx.x >> 5);        // wave id < EE*TT
    int lane = threadIdx.x & 31;
    int e = gid >> 2, t = gid & 3;
    int r = row[e], c = col[e];
    float w = re[(size_t)t * EE + e] / ssum[t * NN + r];
    const float* hv  = hw + ((size_t)t * NN + c) * DD;
    const float* vgt = vg + t * DD;
    float* o = out + ((size_t)t * NN + r) * DD;
    atomicAdd(&o[lane],      w * vgt[lane]      * hv[lane]);
    atomicAdd(&o[lane + 32], w * vgt[lane + 32] * hv[lane + 32]);
}

extern "C" void kernel_launch(void* const* d_in, const int* in_sizes, int n_in,
                              void* d_out, int out_size, void* d_ws, size_t ws_size,
                              hipStream_t stream) {
    const float* x     = (const float*)d_in[0];   // N x D
    const float* efeat = (const float*)d_in[1];   // T x D
    const float* theta = (const float*)d_in[2];   // L x T x 3D
    const float* wr    = (const float*)d_in[3];   // L x D x D
    const float* we    = (const float*)d_in[4];   // L x D x D
    const int*   eidx  = (const int*)d_in[5];     // 2 x E
    const int*   etype = (const int*)d_in[6];     // E
    const int* row = eidx;
    const int* col = eidx + EE;
    float* out = (float*)d_out;                   // T x N x D

    // workspace layout
    float* ws = (float*)d_ws;
    const size_t HND = (size_t)TT * NN * DD;      // 4M floats
    float*    hA     = ws;                        // T*N*D
    float*    hB     = hA + HND;                  // T*N*D
    float*    hw     = hB + HND;                  // T*N*D
    float*    ai     = hw + HND;                  // T*N
    float*    aj     = ai + (size_t)TT * NN;      // T*N
    unsigned* smax   = (unsigned*)(aj + (size_t)TT * NN); // T*N
    float*    ssum   = (float*)(smax + (size_t)TT * NN);  // T*N
    float*    re     = ssum + (size_t)TT * NN;    // T*E
    float*    efA    = re + (size_t)TT * EE;      // T*D
    float*    efB    = efA + TT * DD;             // T*D
    float*    g      = efB + TT * DD;             // T*T
    float*    vg     = g + TT * TT;               // T*D

    // init: broadcast x into h, copy edge features
    k_bcast_h<<<(int)(HND / 256), 256, 0, stream>>>(x, hA);
    k_copy256<<<1, 256, 0, stream>>>(efeat, efA);

    for (int i = 0; i < LL; ++i) {
        const float* h_in  = (i == 0) ? hA : hB;
        float*       h_out = (i == LL - 1) ? out : hB;
        const float* ef_in  = (i == 0) ? efA : efB;
        float*       ef_out = (i == 0) ? efB : efA;
        const float* th_i = theta + (size_t)i * TT * 3 * DD;

        // small dense prep (g uses pre-update ef, per the reference ordering)
        k_prep<<<1, 256, 0, stream>>>(th_i, ef_in, wr + (size_t)i * DD * DD, g, ef_out, vg);

        // per-node attention scores
        k_node_scores<<<TT * NN / 256, 256, 0, stream>>>(h_in, th_i, ai, aj);

        // hw = h @ we via WMMA f32 16x16x4
        k_gemm_wmma<<<dim3(NN / 16, TT), 128, 0, stream>>>(h_in, we + (size_t)i * DD * DD, hw);

        // zero segment-max (0u is the identity of the ordered encoding) + segment-sum + output
        k_fill0<<<(2 * TT * NN) / 256, 256, 0, stream>>>((float*)smax, 2 * TT * NN);
        k_fill0<<<(int)(HND / 256), 256, 0, stream>>>(h_out, (int)HND);

        // scatter softmax
        k_edge_r<<<EE / 256, 256, 0, stream>>>(row, col, etype, g, ai, aj, re, smax);
        k_edge_exp<<<EE / 256, 256, 0, stream>>>(row, smax, re, ssum);

        // weighted aggregation into h_out
        k_agg<<<(EE * TT) / 8, 256, 0, stream>>>(row, col, re, ssum, vg, hw, h_out);

        if (i < LL - 1) {
            k_relu<<<(int)(HND / 256), 256, 0, stream>>>(h_out, (int)HND);
            k_relu<<<1, 256, 0, stream>>>(ef_out, TT * DD);
        }
    }
}